// QuickShiftGroups_2851858284929
// MI455X (gfx1250) — compile-verified
//
#include <hip/hip_runtime.h>

typedef float v2f __attribute__((ext_vector_type(2)));
typedef float v8f __attribute__((ext_vector_type(8)));

#define H 128
#define W 128
#define NPIX (H * W)          // 16384 per image
#define BATCH 2
#define NTOT (BATCH * NPIX)   // 32768
#define KW 30                 // quickshift window radius
#define GR 20                 // gaussian radius (4*sigma+0.5)
#define GLEN 41
#define INV 0.005f            // 0.5 / kernel_size^2
#define MAXD2 400.0f          // max_dist^2
#define MAXSEG 40

// ---------------------------------------------------------------- S matrix
// S[i][j] = sum over taps k of g[k] where symmetric-reflected(i-GR+k) == j
__global__ void build_S_kernel(float* __restrict__ S) {
    int t = blockIdx.x * blockDim.x + threadIdx.x;
    if (t >= H * H) return;
    int i = t >> 7, j = t & 127;
    float gsum = 0.f;
    for (int k = 0; k < GLEN; ++k) {
        float xk = (float)(k - GR) * (1.0f / 5.0f);
        gsum += __expf(-0.5f * xk * xk);
    }
    float acc = 0.f;
    for (int k = 0; k < GLEN; ++k) {
        int idx = i - GR + k;
        if (idx < 0) idx = -idx - 1;           // symmetric pad
        else if (idx >= H) idx = 2 * H - 1 - idx;
        if (idx == j) {
            float xk = (float)(k - GR) * (1.0f / 5.0f);
            acc += __expf(-0.5f * xk * xk) / gsum;
        }
    }
    S[i * H + j] = acc;
}

// ---------------------------------------------------------------- rgb2lab
__global__ void rgb2lab_kernel(const float* __restrict__ x, float* __restrict__ lab) {
    int i = blockIdx.x * blockDim.x + threadIdx.x;
    if (i >= NTOT) return;
    int b = i >> 14, p = i & (NPIX - 1);
    const float* X = x + b * 3 * NPIX;
    float r = X[p], g = X[NPIX + p], bl = X[2 * NPIX + p];
    float rl = r > 0.04045f ? powf((r + 0.055f) / 1.055f, 2.4f) : r * (1.0f / 12.92f);
    float gl = g > 0.04045f ? powf((g + 0.055f) / 1.055f, 2.4f) : g * (1.0f / 12.92f);
    float bb = bl > 0.04045f ? powf((bl + 0.055f) / 1.055f, 2.4f) : bl * (1.0f / 12.92f);
    float Xc = (0.412453f * rl + 0.357580f * gl + 0.180423f * bb) * (1.0f / 0.95047f);
    float Yc = (0.212671f * rl + 0.715160f * gl + 0.072169f * bb);
    float Zc = (0.019334f * rl + 0.119193f * gl + 0.950227f * bb) * (1.0f / 1.08883f);
    float fx = Xc > 0.008856f ? cbrtf(fmaxf(Xc, 1e-8f)) : 7.787f * Xc + 16.0f / 116.0f;
    float fy = Yc > 0.008856f ? cbrtf(fmaxf(Yc, 1e-8f)) : 7.787f * Yc + 16.0f / 116.0f;
    float fz = Zc > 0.008856f ? cbrtf(fmaxf(Zc, 1e-8f)) : 7.787f * Zc + 16.0f / 116.0f;
    float* L = lab + b * 3 * NPIX;
    L[p]            = 116.f * fy - 16.f;
    L[NPIX + p]     = 500.f * (fx - fy);
    L[2 * NPIX + p] = 200.f * (fy - fz);
}

// ---------------------------------------------------------------- WMMA GEMM
// TRANSB=false: out = S * in        (row smoothing,  A = S,  B = image)
// TRANSB=true : out = in * S^T      (col smoothing,  A = image, B = S^T)
// One wave32 per 16x16 output tile; K loop of 32 V_WMMA_F32_16X16X4_F32.
template <bool TRANSB>
__global__ void smooth_gemm_kernel(const float* __restrict__ S,
                                   const float* __restrict__ in,
                                   float* __restrict__ out) {
    const int tn = blockIdx.x, tm = blockIdx.y, z = blockIdx.z;
    const int lane = threadIdx.x;
    const int half = lane >> 4;      // 0: lanes 0-15, 1: lanes 16-31
    const int lm   = lane & 15;
    const float* A = TRANSB ? (in + z * NPIX) : S;
    const float* B = TRANSB ? S : (in + z * NPIX);
    float* O = out + z * NPIX;
    const int arow = tm * 16 + lm;
    const int bcol = tn * 16 + lm;
    v8f c = {};
    for (int k0 = 0; k0 < 128; k0 += 4) {
        const int kk = k0 + 2 * half;   // lanes 16-31 hold K+2,K+3 (ISA 7.12.2)
        v2f a, b;
        a.x = A[arow * 128 + kk];
        a.y = A[arow * 128 + kk + 1];
        if (TRANSB) {                   // B[k][n] = S[n][k]
            b.x = B[bcol * 128 + kk];
            b.y = B[bcol * 128 + kk + 1];
        } else {
            b.x = B[kk * 128 + bcol];
            b.y = B[(kk + 1) * 128 + bcol];
        }
        c = __builtin_amdgcn_wmma_f32_16x16x4_f32(
            /*neg_a=*/false, a, /*neg_b=*/false, b,
            /*c_mod=*/(short)0, c, /*reuse_a=*/false, /*reuse_b=*/false);
    }
    // C/D layout: VGPR v -> row v (lanes 0-15) / row v+8 (lanes 16-31)
    for (int v = 0; v < 8; ++v)
        O[(tm * 16 + v + 8 * half) * 128 + (tn * 16 + lm)] = c[v];
}

// ---------------------------------------------------------------- pack HWC4
__global__ void pack_kernel(const float* __restrict__ feat, float4* __restrict__ f4) {
    int i = blockIdx.x * blockDim.x + threadIdx.x;
    if (i >= NTOT) return;
    int b = i >> 14, p = i & (NPIX - 1);
    const float* F = feat + b * 3 * NPIX;
    f4[i] = make_float4(F[p], F[NPIX + p], F[2 * NPIX + p], 0.f);
}

// ---------------------------------------------------------------- density
__global__ void density_kernel(const float4* __restrict__ f4, float* __restrict__ dens) {
    int i = blockIdx.x * blockDim.x + threadIdx.x;
    if (i >= NTOT) return;
    int b = i >> 14, p = i & (NPIX - 1);
    int y = p >> 7, x = p & 127;
    const float4* F = f4 + b * NPIX;
    float4 f = F[p];
    float acc = 0.f;
    for (int dy = -KW; dy <= KW; ++dy) {
        int ny = y + dy;
        if ((unsigned)ny >= (unsigned)H) continue;
        float dy2 = (float)(dy * dy);
        const float4* row = F + ny * W;
        for (int dx = -KW; dx <= KW; ++dx) {
            int nx = x + dx;
            if ((unsigned)nx >= (unsigned)W) continue;
            float4 nb = row[nx];
            float d = (f.x - nb.x) * (f.x - nb.x) + (f.y - nb.y) * (f.y - nb.y) +
                      (f.z - nb.z) * (f.z - nb.z) + dy2 + (float)(dx * dx);
            acc += __expf(-d * INV);
        }
    }
    dens[i] = acc;
}

// ---------------------------------------------------------------- parent
__global__ void parent_kernel(const float4* __restrict__ f4,
                              const float* __restrict__ dens,
                              int* __restrict__ parent) {
    int i = blockIdx.x * blockDim.x + threadIdx.x;
    if (i >= NTOT) return;
    int b = i >> 14, p = i & (NPIX - 1);
    int y = p >> 7, x = p & 127;
    const float4* F = f4 + b * NPIX;
    const float* D  = dens + b * NPIX;
    float4 f = F[p];
    float myd = D[p];
    float best_d = __builtin_inff();
    int best_p = p;
    // row-major scan order: strict '<' keeps first winner on ties (matches ref)
    for (int dy = -KW; dy <= KW; ++dy) {
        int ny = y + dy;
        if ((unsigned)ny >= (unsigned)H) continue;
        float dy2 = (float)(dy * dy);
        for (int dx = -KW; dx <= KW; ++dx) {
            int nx = x + dx;
            if ((unsigned)nx >= (unsigned)W) continue;
            int np_ = ny * W + nx;
            float nd = D[np_];
            if (nd > myd) {
                float4 nb = F[np_];
                float d = (f.x - nb.x) * (f.x - nb.x) + (f.y - nb.y) * (f.y - nb.y) +
                          (f.z - nb.z) * (f.z - nb.z) + dy2 + (float)(dx * dx);
                if (d < best_d) { best_d = d; best_p = np_; }
            }
        }
    }
    if (best_d > MAXD2) best_p = p;
    parent[i] = best_p;
}

// ---------------------------------------------------------------- root chase
__global__ void root_kernel(const int* __restrict__ parent, int* __restrict__ root) {
    int i = blockIdx.x * blockDim.x + threadIdx.x;
    if (i >= NTOT) return;
    int b = i >> 14;
    const int* P = parent + b * NPIX;
    int p = i & (NPIX - 1);
    // parent density strictly increases along chain -> acyclic, terminates at self-root
    while (P[p] != p) p = P[p];
    root[i] = p;
}

__global__ void clear_kernel(int* __restrict__ a) {
    int i = blockIdx.x * blockDim.x + threadIdx.x;
    if (i < NTOT) a[i] = 0;
}

__global__ void mark_kernel(const int* __restrict__ root, int* __restrict__ present) {
    int i = blockIdx.x * blockDim.x + threadIdx.x;
    if (i >= NTOT) return;
    present[(i >> 14) * NPIX + root[i]] = 1;
}

// inclusive prefix sum of present[] per image; 1 block (256 thr) per image
__global__ void scan_kernel(const int* __restrict__ present, int* __restrict__ cum) {
    __shared__ int sums[256];
    int b = blockIdx.x, t = threadIdx.x;
    const int* P = present + b * NPIX;
    int* C = cum + b * NPIX;
    const int chunk = NPIX / 256;  // 64
    int s = 0;
    for (int j = t * chunk; j < (t + 1) * chunk; ++j) s += P[j];
    sums[t] = s;
    __syncthreads();
    for (int off = 1; off < 256; off <<= 1) {
        int v = (t >= off) ? sums[t - off] : 0;
        __syncthreads();
        sums[t] += v;
        __syncthreads();
    }
    int run = sums[t] - s;  // exclusive prefix for this chunk
    for (int j = t * chunk; j < (t + 1) * chunk; ++j) { run += P[j]; C[j] = run; }
}

__global__ void label_kernel(const int* __restrict__ root, const int* __restrict__ cum,
                             int* __restrict__ out) {
    int i = blockIdx.x * blockDim.x + threadIdx.x;
    if (i >= NTOT) return;
    int lbl = cum[(i >> 14) * NPIX + root[i]] - 1;
    out[i] = lbl > (MAXSEG - 1) ? (MAXSEG - 1) : lbl;
}

// ---------------------------------------------------------------- launcher
extern "C" void kernel_launch(void* const* d_in, const int* in_sizes, int n_in,
                              void* d_out, int out_size, void* d_ws, size_t ws_size,
                              hipStream_t stream) {
    (void)in_sizes; (void)n_in; (void)out_size; (void)ws_size;
    const float* x = (const float*)d_in[0];
    int* out = (int*)d_out;  // reference output dtype is int32

    char* ws = (char*)d_ws;
    const size_t LABSZ = (size_t)BATCH * 3 * NPIX * sizeof(float);  // 393216
    float*  lab    = (float*)(ws);
    float*  tmp    = (float*)(ws + LABSZ);
    float*  feat   = (float*)(ws + 2 * LABSZ);
    float*  S      = (float*)(ws + 3 * LABSZ);
    float4* f4     = (float4*)(ws + 3 * LABSZ + (size_t)H * H * sizeof(float));
    char*   after4 = (char*)(f4 + NTOT);
    float*  dens   = (float*)(after4);
    int*    parent = (int*)(after4 + NTOT * sizeof(float));
    int*    root   = (int*)(after4 + 2 * NTOT * sizeof(float));
    int*    pres   = (int*)(after4 + 3 * NTOT * sizeof(float));
    int*    cum    = (int*)(after4 + 4 * NTOT * sizeof(float));

    const int TPB = 256;
    build_S_kernel<<<(H * H + TPB - 1) / TPB, TPB, 0, stream>>>(S);
    rgb2lab_kernel<<<(NTOT + TPB - 1) / TPB, TPB, 0, stream>>>(x, lab);

    dim3 gtile(W / 16, H / 16, BATCH * 3);   // 8 x 8 x 6 tiles, one wave each
    smooth_gemm_kernel<false><<<gtile, 32, 0, stream>>>(S, lab, tmp);   // rows: S * X
    smooth_gemm_kernel<true ><<<gtile, 32, 0, stream>>>(S, tmp, feat);  // cols: X * S^T

    pack_kernel<<<(NTOT + TPB - 1) / TPB, TPB, 0, stream>>>(feat, f4);
    density_kernel<<<(NTOT + TPB - 1) / TPB, TPB, 0, stream>>>(f4, dens);
    parent_kernel<<<(NTOT + TPB - 1) / TPB, TPB, 0, stream>>>(f4, dens, parent);
    root_kernel<<<(NTOT + TPB - 1) / TPB, TPB, 0, stream>>>(parent, root);
    clear_kernel<<<(NTOT + TPB - 1) / TPB, TPB, 0, stream>>>(pres);
    mark_kernel<<<(NTOT + TPB - 1) / TPB, TPB, 0, stream>>>(root, pres);
    scan_kernel<<<BATCH, 256, 0, stream>>>(pres, cum);
    label_kernel<<<(NTOT + TPB - 1) / TPB, TPB, 0, stream>>>(root, cum, out);
}